// LSTM_47236050322023
// MI455X (gfx1250) — compile-verified
//
#include <hip/hip_runtime.h>
#include <math.h>

// LSTM forward, B=64, T=512, D=1024, fp32 (PyTorch gate order i,f,g,o).
// Persistent-kernel design for MI455X:
//  - one launch; the 512 sequential timesteps run inside the kernel with a
//    device-wide sense barrier (atomics in d_ws) -> no per-step launch cost.
//  - 256 blocks = 64 d-tiles x 4 batch m-tiles; 4 wave32/block split K,
//    each wave holds 4 gate accumulators (V_WMMA_F32_16X16X4_F32, exact fp32).
//  - weights (32 MB fp32) + h/c state live in the 192 MB L2 across all steps;
//    HBM only streams x once. Next-step x rows are prefetched
//    (global_prefetch_b8) while this step finishes.

#define LSTM_D   1024
#define LSTM_B   64
#define LSTM_T   512
#define NBLK     256          // (LSTM_B/16) * (LSTM_D/16) = 4 * 64
#define NTHR     128          // 4 wave32
#define KSLICE   256          // 1024 / 4 waves, per GEMM phase

typedef float v2f __attribute__((ext_vector_type(2)));
typedef float v8f __attribute__((ext_vector_type(8)));

__global__ __launch_bounds__(256) void lstm_init_kernel(float* p, int n) {
    int i = blockIdx.x * blockDim.x + threadIdx.x;
    if (i < n) p[i] = 0.0f;                  // zeros c, h0, h1, and barrier words
}

__global__ __launch_bounds__(NTHR) void lstm_persist_kernel(
    const float* __restrict__ x,    // [B, T, D]
    const float* __restrict__ Wih,  // [4D, D]
    const float* __restrict__ Whh,  // [4D, D]
    const float* __restrict__ bih,  // [4D]
    const float* __restrict__ bhh,  // [4D]
    float* __restrict__ h0,         // [B, D] ping
    float* __restrict__ h1,         // [B, D] pong
    float* __restrict__ cSt,        // [B, D]
    float* __restrict__ out,        // [B, D] final h
    unsigned* __restrict__ barCnt,
    unsigned* __restrict__ barRel)
{
    __shared__ float part[3][4][16][16];   // waves 1..3 partial gate tiles (12 KB)
    __shared__ float gfin[4][16][16];      // reduced gate tiles (4 KB)

    const int lane  = threadIdx.x & 31;
    const int wave  = threadIdx.x >> 5;        // k-slice id 0..3
    const int mTile = blockIdx.x & 3;          // batch tile (16 rows)
    const int dTile = blockIdx.x >> 2;         // d tile (16 cols)
    const int d0    = dTile * 16;

    // WMMA f32 16x16x4 per-lane fragment addressing:
    //   A (16x4): lane l -> A[l&15, kh], A[l&15, kh+1], kh = 2*(l>>4)
    //   B (4x16): lane l -> B[kh, l&15], B[kh+1, l&15]; B[k,n] = W[nGlob, k]
    const int mrow  = lane & 15;
    const int kh    = (lane >> 4) * 2;
    const int kBase = wave * KSLICE;

    const size_t TD = (size_t)LSTM_T * LSTM_D;
    const float* xRowBase = x + (size_t)(mTile * 16 + mrow) * TD + kBase + kh;

    const float* wi0 = Wih + (size_t)(0 * LSTM_D + d0 + mrow) * LSTM_D + kBase + kh;
    const float* wi1 = Wih + (size_t)(1 * LSTM_D + d0 + mrow) * LSTM_D + kBase + kh;
    const float* wi2 = Wih + (size_t)(2 * LSTM_D + d0 + mrow) * LSTM_D + kBase + kh;
    const float* wi3 = Wih + (size_t)(3 * LSTM_D + d0 + mrow) * LSTM_D + kBase + kh;
    const float* wh0 = Whh + (size_t)(0 * LSTM_D + d0 + mrow) * LSTM_D + kBase + kh;
    const float* wh1 = Whh + (size_t)(1 * LSTM_D + d0 + mrow) * LSTM_D + kBase + kh;
    const float* wh2 = Whh + (size_t)(2 * LSTM_D + d0 + mrow) * LSTM_D + kBase + kh;
    const float* wh3 = Whh + (size_t)(3 * LSTM_D + d0 + mrow) * LSTM_D + kBase + kh;

    const size_t hOff = (size_t)(mTile * 16 + mrow) * LSTM_D + kBase + kh;

    for (int t = 0; t < LSTM_T; ++t) {
        const float* xr = xRowBase + (size_t)t * LSTM_D;
        const float* hr = ((t & 1) ? h1 : h0) + hOff;

        v8f a0 = {}, a1 = {}, a2 = {}, a3 = {};   // gate i,f,g,o partial tiles

        // ---- x_t @ W_ih^T, this wave's K slice ----
        #pragma unroll 4
        for (int k0 = 0; k0 < KSLICE; k0 += 4) {
            v2f av = *(const v2f*)(xr + k0);       // A reused across 4 gates
            v2f b0 = *(const v2f*)(wi0 + k0);
            v2f b1 = *(const v2f*)(wi1 + k0);
            v2f b2 = *(const v2f*)(wi2 + k0);
            v2f b3 = *(const v2f*)(wi3 + k0);
            a0 = __builtin_amdgcn_wmma_f32_16x16x4_f32(false, av, false, b0, (short)0, a0, false, false);
            a1 = __builtin_amdgcn_wmma_f32_16x16x4_f32(false, av, false, b1, (short)0, a1, false, false);
            a2 = __builtin_amdgcn_wmma_f32_16x16x4_f32(false, av, false, b2, (short)0, a2, false, false);
            a3 = __builtin_amdgcn_wmma_f32_16x16x4_f32(false, av, false, b3, (short)0, a3, false, false);
        }
        // ---- h_{t-1} @ W_hh^T, this wave's K slice ----
        #pragma unroll 4
        for (int k0 = 0; k0 < KSLICE; k0 += 4) {
            v2f av = *(const v2f*)(hr + k0);
            v2f b0 = *(const v2f*)(wh0 + k0);
            v2f b1 = *(const v2f*)(wh1 + k0);
            v2f b2 = *(const v2f*)(wh2 + k0);
            v2f b3 = *(const v2f*)(wh3 + k0);
            a0 = __builtin_amdgcn_wmma_f32_16x16x4_f32(false, av, false, b0, (short)0, a0, false, false);
            a1 = __builtin_amdgcn_wmma_f32_16x16x4_f32(false, av, false, b1, (short)0, a1, false, false);
            a2 = __builtin_amdgcn_wmma_f32_16x16x4_f32(false, av, false, b2, (short)0, a2, false, false);
            a3 = __builtin_amdgcn_wmma_f32_16x16x4_f32(false, av, false, b3, (short)0, a3, false, false);
        }

        // ---- cross-wave K reduction through LDS ----
        // C/D layout: VGPR v: lanes 0-15 -> M=v, lanes 16-31 -> M=v+8; N = lane&15.
        const int nCol = lane & 15;
        const int mOff = (lane >> 4) * 8;
        if (wave != 0) {
            #pragma unroll
            for (int v = 0; v < 8; ++v) {
                part[wave - 1][0][mOff + v][nCol] = a0[v];
                part[wave - 1][1][mOff + v][nCol] = a1[v];
                part[wave - 1][2][mOff + v][nCol] = a2[v];
                part[wave - 1][3][mOff + v][nCol] = a3[v];
            }
        }
        __syncthreads();
        if (wave == 0) {
            #pragma unroll
            for (int v = 0; v < 8; ++v) {
                const int r = mOff + v;
                gfin[0][r][nCol] = a0[v] + part[0][0][r][nCol] + part[1][0][r][nCol] + part[2][0][r][nCol];
                gfin[1][r][nCol] = a1[v] + part[0][1][r][nCol] + part[1][1][r][nCol] + part[2][1][r][nCol];
                gfin[2][r][nCol] = a2[v] + part[0][2][r][nCol] + part[1][2][r][nCol] + part[2][2][r][nCol];
                gfin[3][r][nCol] = a3[v] + part[0][3][r][nCol] + part[1][3][r][nCol] + part[2][3][r][nCol];
            }
        }
        __syncthreads();

        // ---- pointwise LSTM cell update for this (mTile, dTile) tile ----
        float* hOut = (t == LSTM_T - 1) ? out : ((t & 1) ? h0 : h1);
        for (int e = threadIdx.x; e < 256; e += NTHR) {
            const int r   = e >> 4;
            const int col = e & 15;
            const int gr  = mTile * 16 + r;
            const int d   = d0 + col;
            const float gi = gfin[0][r][col] + bih[d]                + bhh[d];
            const float gf = gfin[1][r][col] + bih[LSTM_D + d]       + bhh[LSTM_D + d];
            const float gg = gfin[2][r][col] + bih[2 * LSTM_D + d]   + bhh[2 * LSTM_D + d];
            const float go = gfin[3][r][col] + bih[3 * LSTM_D + d]   + bhh[3 * LSTM_D + d];
            const float ig = 1.0f / (1.0f + expf(-gi));
            const float fg = 1.0f / (1.0f + expf(-gf));
            const float og = 1.0f / (1.0f + expf(-go));
            const size_t idx = (size_t)gr * LSTM_D + d;
            const float cN = fg * cSt[idx] + ig * tanhf(gg);
            cSt[idx]  = cN;
            hOut[idx] = og * tanhf(cN);
        }

        // Warm L2/WGP path for next step's x rows while we wait at the barrier.
        if (t + 1 < LSTM_T)
            __builtin_prefetch(xRowBase + (size_t)(t + 1) * LSTM_D, 0, 3);

        // ---- device-wide barrier (monotonic count + release counter) ----
        __threadfence();
        __syncthreads();
        if (threadIdx.x == 0) {
            const unsigned arrive = atomicAdd(barCnt, 1u);
            if (arrive == (unsigned)(t * NBLK + (NBLK - 1))) {
                atomicExch(barRel, (unsigned)(t + 1));     // last arrival releases
            } else {
                while (atomicAdd(barRel, 0u) <= (unsigned)t)
                    __builtin_amdgcn_s_sleep(8);
            }
        }
        __syncthreads();
        __threadfence();
    }
}

extern "C" void kernel_launch(void* const* d_in, const int* in_sizes, int n_in,
                              void* d_out, int out_size, void* d_ws, size_t ws_size,
                              hipStream_t stream) {
    (void)in_sizes; (void)n_in; (void)out_size; (void)ws_size;
    const float* x   = (const float*)d_in[0];
    const float* Wih = (const float*)d_in[1];
    const float* Whh = (const float*)d_in[2];
    const float* bih = (const float*)d_in[3];
    const float* bhh = (const float*)d_in[4];
    float* out = (float*)d_out;

    float* ws  = (float*)d_ws;
    float* cSt = ws;                               // [B, D]
    float* h0  = ws + 1 * LSTM_B * LSTM_D;         // [B, D] ping
    float* h1  = ws + 2 * LSTM_B * LSTM_D;         // [B, D] pong
    unsigned* bar = (unsigned*)(ws + 3 * LSTM_B * LSTM_D);  // [2]: count, release

    // Zero c, h0, h1 and the barrier words (state must be rebuilt every call).
    const int nz = 3 * LSTM_B * LSTM_D + 2;
    lstm_init_kernel<<<(nz + 255) / 256, 256, 0, stream>>>(ws, nz);

    lstm_persist_kernel<<<NBLK, NTHR, 0, stream>>>(
        x, Wih, Whh, bih, bhh, h0, h1, cSt, out, bar, bar + 1);
}